// BWGNN_Hetero_36636071035259
// MI455X (gfx1250) — compile-verified
//
#include <hip/hip_runtime.h>

#define N_NODES 50000
#define N_EDGES 800000
#define IN_FEATS 128
#define HDIM 64
#define LSTRIDE 68   // LDS row stride (floats): conflict-free + 8B aligned

typedef __attribute__((ext_vector_type(2))) float v2f;
typedef __attribute__((ext_vector_type(8))) float v8f;

__device__ __forceinline__ float leaky_f(float x) { return x > 0.0f ? x : 0.01f * x; }

// ---------------- elementwise helpers ----------------
__global__ void zero_f32(float* __restrict__ p, int n) {
    int i = blockIdx.x * blockDim.x + threadIdx.x;
    if (i < n) p[i] = 0.0f;
}

// out = 2*b3[col]  (both relation branches add +b3; fold into the init)
__global__ void init_out_bias2(float* __restrict__ out, const float* __restrict__ b3, int n) {
    int i = blockIdx.x * blockDim.x + threadIdx.x;
    if (i < n) out[i] = 2.0f * b3[i & 63];
}

__global__ void leaky_inplace(float* __restrict__ p, int n) {
    int i = blockIdx.x * blockDim.x + threadIdx.x;
    if (i < n) p[i] = leaky_f(p[i]);
}

// deg[dst[e]] += 1
__global__ void degree_kernel(const int* __restrict__ dst, float* __restrict__ deg) {
    int e = blockIdx.x * blockDim.x + threadIdx.x;
    if (e < N_EDGES) atomicAdd(&deg[dst[e]], 1.0f);
}

// deg -> clip(deg,1)^(-1/2), in place
__global__ void dinv_kernel(float* __restrict__ deg) {
    int i = blockIdx.x * blockDim.x + threadIdx.x;
    if (i < N_NODES) deg[i] = rsqrtf(fmaxf(deg[i], 1.0f));
}

// agg[dst] += feat[src] * dinv[src]; one thread per edge (indices read once)
__global__ void scatter_edges(const float4* __restrict__ feat, const int* __restrict__ src,
                              const int* __restrict__ dst, const float* __restrict__ dinv,
                              float* __restrict__ agg) {
    int e = blockIdx.x * blockDim.x + threadIdx.x;
    if (e >= N_EDGES) return;
    int s = src[e], d = dst[e];
    float w = dinv[s];
    const float4* fs = feat + (size_t)s * 16;
    float* a = agg + (size_t)d * 64;
#pragma unroll 4
    for (int c = 0; c < 16; ++c) {
        float4 v = fs[c];
        atomicAdd(a + c * 4 + 0, v.x * w);
        atomicAdd(a + c * 4 + 1, v.y * w);
        atomicAdd(a + c * 4 + 2, v.z * w);
        atomicAdd(a + c * 4 + 3, v.w * w);
    }
}

// Wc_k = sum_t theta[t][k] * W3_block_t   (theta0=[3,-3,.75], theta1=[0,3,-1.5], theta2=[0,0,.75])
__global__ void build_wc(const float* __restrict__ W3, float* __restrict__ Wc) {
    int i = blockIdx.x * blockDim.x + threadIdx.x;     // 3*64*64
    if (i >= 3 * 64 * 64) return;
    int k = i >> 12, ij = i & 4095;
    float b0 = W3[ij], b1 = W3[4096 + ij], b2 = W3[8192 + ij];
    float v;
    if (k == 0)      v =  3.0f  * b0;
    else if (k == 1) v = -3.0f  * b0 + 3.0f * b1;
    else             v =  0.75f * b0 - 1.5f * b1 + 0.75f * b2;
    Wc[i] = v;
}

// ---------------- fused 2-layer MLP via WMMA + LDS staging ----------------
// Feat = leaky(leaky(X@W1 + b1) @ W2 + b2); one wave32 owns a 16x64 output block.
__global__ void fused_mlp(const float* __restrict__ X, const float* __restrict__ W1,
                          const float* __restrict__ b1, const float* __restrict__ W2,
                          const float* __restrict__ b2, float* __restrict__ Feat) {
    __shared__ float lds[8 * 16 * LSTRIDE];            // 8 waves/block, 16x64 tile each
    int wave = (blockIdx.x * blockDim.x + threadIdx.x) >> 5;
    int wid  = threadIdx.x >> 5;
    int lane = threadIdx.x & 31;
    if (wave >= N_NODES / 16) return;                  // wave-uniform guard
    int half = lane >> 4, l16 = lane & 15;
    float* L = lds + wid * 16 * LSTRIDE;

    // ---- layer 1: X(16x128) @ W1(128x64) ----
    const float* Arow = X + (size_t)(wave * 16 + l16) * IN_FEATS;
    v8f acc[4] = {};
#pragma unroll 2
    for (int k0 = 0; k0 < IN_FEATS; k0 += 4) {
        int kk = k0 + 2 * half;                        // v0: K0/K2, v1: K1/K3 per lane half
        __builtin_prefetch((const void*)(Arow + k0 + 64), 0, 0);
        v2f a;
        a.x = Arow[kk];
        a.y = Arow[kk + 1];
#pragma unroll
        for (int t = 0; t < 4; ++t) {
            v2f b;
            b.x = W1[(size_t)kk * 64 + t * 16 + l16];
            b.y = W1[(size_t)(kk + 1) * 64 + t * 16 + l16];
            acc[t] = __builtin_amdgcn_wmma_f32_16x16x4_f32(false, a, false, b,
                                                           (short)0, acc[t], false, false);
        }
    }
    // bias + leaky, stage to LDS in row-major (C layout -> [row][col])
#pragma unroll
    for (int t = 0; t < 4; ++t) {
        float bv = b1[t * 16 + l16];
#pragma unroll
        for (int r = 0; r < 8; ++r)
            L[(r + 8 * half) * LSTRIDE + t * 16 + l16] = leaky_f(acc[t][r] + bv);
    }
    // same-wave LDS ops are in-order; fence to be explicit and pin compiler ordering
    asm volatile("s_wait_dscnt 0x0" ::: "memory");

    // ---- layer 2: H(16x64) @ W2(64x64), A-fragments read from LDS ----
    const float* Lrow = L + l16 * LSTRIDE;
    v8f acc2[4] = {};
#pragma unroll 2
    for (int k0 = 0; k0 < HDIM; k0 += 4) {
        int kk = k0 + 2 * half;
        v2f a;
        a.x = Lrow[kk];
        a.y = Lrow[kk + 1];
#pragma unroll
        for (int t = 0; t < 4; ++t) {
            v2f b;
            b.x = W2[(size_t)kk * 64 + t * 16 + l16];
            b.y = W2[(size_t)(kk + 1) * 64 + t * 16 + l16];
            acc2[t] = __builtin_amdgcn_wmma_f32_16x16x4_f32(false, a, false, b,
                                                            (short)0, acc2[t], false, false);
        }
    }
#pragma unroll
    for (int t = 0; t < 4; ++t) {
        float bv = b2[t * 16 + l16];
#pragma unroll
        for (int r = 0; r < 8; ++r)
            Feat[(size_t)(wave * 16 + r + 8 * half) * 64 + t * 16 + l16] =
                leaky_f(acc2[t][r] + bv);
    }
}

// ---------------- polynomial-term GEMM: out += A' @ Wc  ----------------
// doUpdate==0: A' = Feat (read only)
// doUpdate==1: A' = Feat - Agg*dinv[row]; Feat <- A'; Agg <- 0 (fused propagation update)
__global__ void gemm_poly(float* __restrict__ Feat, float* __restrict__ Agg,
                          const float* __restrict__ dinv, const float* __restrict__ B,
                          float* __restrict__ out, int doUpdate) {
    int wave = (blockIdx.x * blockDim.x + threadIdx.x) >> 5;
    int lane = threadIdx.x & 31;
    if (wave >= N_NODES / 16) return;                  // wave-uniform guard
    int half = lane >> 4, l16 = lane & 15;
    int row = wave * 16 + l16;
    float* Frow = Feat + (size_t)row * 64;
    float* Grow = Agg + (size_t)row * 64;
    float dv = 0.0f;
    if (doUpdate) dv = dinv[row];

    v8f acc[4];
#pragma unroll
    for (int t = 0; t < 4; ++t)
#pragma unroll
        for (int r = 0; r < 8; ++r)
            acc[t][r] = out[(size_t)(wave * 16 + r + 8 * half) * 64 + t * 16 + l16];

#pragma unroll 2
    for (int k0 = 0; k0 < HDIM; k0 += 4) {
        int kk = k0 + 2 * half;
        v2f a;
        if (doUpdate) {
            float f0 = Frow[kk]     - Grow[kk]     * dv;
            float f1 = Frow[kk + 1] - Grow[kk + 1] * dv;
            Frow[kk] = f0; Frow[kk + 1] = f1;          // each element owned by one lane
            Grow[kk] = 0.0f; Grow[kk + 1] = 0.0f;      // pre-zero Agg for next scatter
            a.x = f0; a.y = f1;
        } else {
            a.x = Frow[kk];
            a.y = Frow[kk + 1];
        }
#pragma unroll
        for (int t = 0; t < 4; ++t) {
            v2f b;
            b.x = B[(size_t)kk * 64 + t * 16 + l16];
            b.y = B[(size_t)(kk + 1) * 64 + t * 16 + l16];
            acc[t] = __builtin_amdgcn_wmma_f32_16x16x4_f32(false, a, false, b,
                                                           (short)0, acc[t], false, false);
        }
    }
#pragma unroll
    for (int t = 0; t < 4; ++t)
#pragma unroll
        for (int r = 0; r < 8; ++r)
            out[(size_t)(wave * 16 + r + 8 * half) * 64 + t * 16 + l16] = acc[t][r];
}

// ---------------- host orchestration ----------------
extern "C" void kernel_launch(void* const* d_in, const int* in_sizes, int n_in,
                              void* d_out, int out_size, void* d_ws, size_t ws_size,
                              hipStream_t stream) {
    const float* feat_r1 = (const float*)d_in[0];
    const float* feat_r2 = (const float*)d_in[1];
    const int*   src_r1  = (const int*)d_in[2];
    const int*   dst_r1  = (const int*)d_in[3];
    const int*   src_r2  = (const int*)d_in[4];
    const int*   dst_r2  = (const int*)d_in[5];
    const float* W1_r1   = (const float*)d_in[6];
    const float* b1_r1   = (const float*)d_in[7];
    const float* W2_r1   = (const float*)d_in[8];
    const float* b2_r1   = (const float*)d_in[9];
    const float* W1_r2   = (const float*)d_in[10];
    const float* b1_r2   = (const float*)d_in[11];
    const float* W2_r2   = (const float*)d_in[12];
    const float* b2_r2   = (const float*)d_in[13];
    const float* W3      = (const float*)d_in[14];
    const float* b3      = (const float*)d_in[15];
    float* out = (float*)d_out;
    float* ws  = (float*)d_ws;

    const size_t NF = (size_t)N_NODES * 64;
    float* Feat = ws;                 // N*64 (evolving L^k h)
    float* Agg  = ws + NF;            // N*64
    float* Dinv = ws + 2 * NF;        // N   (deg, then d^-1/2 in place)
    float* Wc   = Dinv + N_NODES;     // 3*64*64 folded W3 blocks

    const int TB = 256;
    auto cdiv = [](long a, long b) { return (int)((a + b - 1) / b); };
    const int waveBlocks = cdiv((long)(N_NODES / 16) * 32, TB);   // one wave per 16 rows

    build_wc<<<cdiv(3 * 4096, TB), TB, 0, stream>>>(W3, Wc);
    init_out_bias2<<<cdiv((long)NF, TB), TB, 0, stream>>>(out, b3, (int)NF);
    zero_f32<<<cdiv((long)NF, TB), TB, 0, stream>>>(Agg, (int)NF);  // only explicit Agg zero

    struct Rel { const float* X; const int* src; const int* dst;
                 const float *W1, *b1, *W2, *b2; };
    Rel rels[2] = { { feat_r1, src_r1, dst_r1, W1_r1, b1_r1, W2_r1, b2_r1 },
                    { feat_r2, src_r2, dst_r2, W1_r2, b1_r2, W2_r2, b2_r2 } };

    for (int r = 0; r < 2; ++r) {
        const Rel& R = rels[r];
        // Feat = h = P0 (fused 2-layer MLP)
        fused_mlp<<<waveBlocks, TB, 0, stream>>>(R.X, R.W1, R.b1, R.W2, R.b2, Feat);
        // degrees -> d^-1/2
        zero_f32<<<cdiv(N_NODES, TB), TB, 0, stream>>>(Dinv, N_NODES);
        degree_kernel<<<cdiv(N_EDGES, TB), TB, 0, stream>>>(R.dst, Dinv);
        dinv_kernel<<<cdiv(N_NODES, TB), TB, 0, stream>>>(Dinv);
        // out += P0 @ Wc0
        gemm_poly<<<waveBlocks, TB, 0, stream>>>(Feat, Agg, Dinv, Wc + 0 * 4096, out, 0);
        // P_k = L P_{k-1}; out += P_k @ Wc_k  (update + Agg-rezero fused into GEMM)
        for (int k = 1; k <= 2; ++k) {
            scatter_edges<<<cdiv(N_EDGES, TB), TB, 0, stream>>>(
                (const float4*)Feat, R.src, R.dst, Dinv, Agg);
            gemm_poly<<<waveBlocks, TB, 0, stream>>>(Feat, Agg, Dinv, Wc + k * 4096, out, 1);
        }
    }
    leaky_inplace<<<cdiv((long)NF, TB), TB, 0, stream>>>(out, (int)NF);
}